// MinimumSpanningMTnDTree_28810640622324
// MI455X (gfx1250) — compile-verified
//
#include <hip/hip_runtime.h>
#include <hip/hip_bf16.h>
#include <stdint.h>

typedef float v2f __attribute__((ext_vector_type(2)));
typedef float v8f __attribute__((ext_vector_type(8)));

namespace {
constexpr int B_ = 4, C_ = 64, H_ = 288, W_ = 432, TEM_ = 6;
constexpr int V_ = H_ * W_;                 // 124416
constexpr int PW = W_ / TEM_;               // 72
constexpr int NR = (H_ - 1) * PW;           // 20664 row edges / phase
constexpr int NC = H_ * (PW - 1);           // 20448 col edges / phase
constexpr int NX = H_ * PW;                 // 20736 cross edges / pair
constexpr int E_ = TEM_ * NR + TEM_ * NC + (TEM_ - 1) * NX; // 350352
constexpr int PHASE_BLK = NR + NC;          // 41112 (weight-order block per phase)
constexpr int CROSS_OFF = TEM_ * PHASE_BLK; // 246672
constexpr int ROWS_END = TEM_ * NR;         // 123984 (index-order)
constexpr int COLS_END = ROWS_END + TEM_ * NC; // 246672
constexpr int ROUNDS = 18;                  // ceil(log2(V)) + 1, as in reference (upper bound)
constexpr int JUMPS  = 18;                  // 2^18 > V: compression upper bound
}

// Weight-order slot -> pixel pair (reference _build_weight_norm2 ordering)
__device__ __forceinline__ void wpair(int e, int& a, int& b) {
  if (e < CROSS_OFF) {
    int i = e / PHASE_BLK, r = e - i * PHASE_BLK;
    if (r < NR) { int h = r / PW, w = r - h * PW; a = h * W_ + i * PW + w; b = a + W_; }
    else { r -= NR; int h = r / (PW - 1), w = r - h * (PW - 1); a = h * W_ + i * PW + w; b = a + 1; }
  } else {
    int t = e - CROSS_OFF; int i = t / NX, r = t - i * NX;
    int h = r / PW, w = r - h * PW; a = h * W_ + i * PW + w; b = a + PW;
  }
}

// Index-order slot -> endpoints (reference _build_index ordering)
__device__ __forceinline__ void ipair(int e, int& u, int& v) {
  if (e < ROWS_END) {
    int i = e / NR, r = e - i * NR; int h = r / PW, w = r - h * PW;
    u = h * W_ + i * PW + w; v = u + W_;
  } else if (e < COLS_END) {
    int t = e - ROWS_END; int i = t / NC, r = t - i * NC;
    int h = r / (PW - 1), w = r - h * (PW - 1);
    u = h * W_ + i * PW + w; v = u + 1;
  } else {
    int t = e - COLS_END; int i = t / NX, r = t - i * NX;
    int h = r / PW, w = r - h * PW;
    u = h * W_ + i * PW + w; v = u + PW;
  }
}

__global__ void setup_uv_kernel(int* __restrict__ u, int* __restrict__ v) {
  int e = blockIdx.x * blockDim.x + threadIdx.x;
  if (e < E_) { int uu, vv; ipair(e, uu, vv); u[e] = uu; v[e] = vv; }
}

__global__ void init_kernel(int* __restrict__ comp, unsigned long long* __restrict__ minkey,
                            unsigned char* __restrict__ chosen, int* __restrict__ out,
                            int* __restrict__ done) {
  int gid = blockIdx.x * blockDim.x + threadIdx.x;
  int stride = gridDim.x * blockDim.x;
  if (gid < B_) done[gid] = 0;
  for (int i = gid; i < B_ * E_; i += stride) {
    if (i < B_ * V_) { comp[i] = i % V_; minkey[i] = ~0ULL; }
    if (i < B_ * (V_ - 1)) out[i] = E_;   // reference pads with E
    chosen[i] = 0;
  }
}

// One wave computes 16 edge weights via the fp32 WMMA Gram trick:
// A (16x4) and B (4x16) share the identical (lane,vgpr)->(m,k) layout, so
// wmma(D, D) accumulates D * D^T; the diagonal is sum_c (x_u - x_v)^2.
__global__ void weight_wmma_kernel(const float* __restrict__ g,
                                   unsigned long long* __restrict__ wkey) {
  int lane = threadIdx.x & 31;
  int wv = blockIdx.x * (blockDim.x >> 5) + (threadIdx.x >> 5);
  int b = blockIdx.y;
  int base = wv * 16;
  if (base >= E_) return;            // wave-uniform: surviving waves keep EXEC all-1s
  int m = lane & 15, h = lane >> 4;
  int e = base + m; int ec = e < E_ ? e : E_ - 1;
  int pa, pb; wpair(ec, pa, pb);
  const float* xa = g + (size_t)b * C_ * V_ + pa;
  const float* xb = g + (size_t)b * C_ * V_ + pb;
  v8f acc = {};
#pragma unroll
  for (int t = 0; t < 16; ++t) {
    int c0 = 4 * t + 2 * h;          // k = vgpr + 2*(lane/16) within the K=4 chunk
    v2f ab;
    ab.x = xa[(size_t)c0 * V_]       - xb[(size_t)c0 * V_];
    ab.y = xa[(size_t)(c0 + 1) * V_] - xb[(size_t)(c0 + 1) * V_];
    acc = __builtin_amdgcn_wmma_f32_16x16x4_f32(false, ab, false, ab,
                                                (short)0, acc, false, false);
  }
  // Diagonal: edge m lives at lane m (m<8, vgpr m) or lane m+16 (m>=8, vgpr m-8)
  int sel = -1;
  if (lane < 8) sel = lane;
  else if (lane >= 16 && lane < 24) sel = lane - 8;
  if (sel >= 0) {
    int k = (lane < 8) ? lane : (lane - 16);
    float val = acc[0];
    if (k == 1) val = acc[1];
    if (k == 2) val = acc[2];
    if (k == 3) val = acc[3];
    if (k == 4) val = acc[4];
    if (k == 5) val = acc[5];
    if (k == 6) val = acc[6];
    if (k == 7) val = acc[7];
    int ee = base + sel;
    if (ee < E_) {
      // (weight_bits, edge_id) lex order == stable-argsort rank order (w >= 0)
      wkey[(size_t)b * E_ + ee] =
          ((unsigned long long)__float_as_uint(val) << 32) | (unsigned)ee;
    }
  }
}

__global__ void relax_kernel(const int* __restrict__ u, const int* __restrict__ v,
                             const unsigned long long* __restrict__ wkey,
                             const int* __restrict__ comp_g,
                             unsigned long long* __restrict__ minkey_g,
                             const int* __restrict__ done) {
  int b = blockIdx.y;
  if (done[b]) return;               // batch fully merged: skip remaining rounds
  int e = blockIdx.x * blockDim.x + threadIdx.x;
  if (e >= E_) return;
  const int* comp = comp_g + (size_t)b * V_;
  int cu = comp[u[e]], cv = comp[v[e]];
  if (cu != cv) {
    unsigned long long k = wkey[(size_t)b * E_ + e];
    unsigned long long* mk = minkey_g + (size_t)b * V_;
    atomicMin(&mk[cu], k);   // GLOBAL_ATOMIC_MIN_U64
    atomicMin(&mk[cv], k);
  }
}

// One workgroup per batch: __syncthreads() is a global sync for that batch's arrays.
__global__ __launch_bounds__(1024)
void merge_kernel(const int* __restrict__ u, const int* __restrict__ v,
                  unsigned long long* __restrict__ minkey_g, int* __restrict__ comp_g,
                  int* __restrict__ parent_g, int* __restrict__ parent2_g,
                  unsigned char* __restrict__ chosen_g, int* __restrict__ done) {
  __shared__ int sAny;
  __shared__ int sChanged;
  int b = blockIdx.x;
  if (done[b]) return;
  unsigned long long* minkey = minkey_g + (size_t)b * V_;
  int* comp = comp_g + (size_t)b * V_;
  int* p0 = parent_g + (size_t)b * V_;
  int* p1 = parent2_g + (size_t)b * V_;
  unsigned char* chosen = chosen_g + (size_t)b * E_;
  int tid = threadIdx.x, T = blockDim.x;

  if (tid == 0) sAny = 0;
  __syncthreads();

  // parent from per-component min edge; mark chosen
  int myAny = 0;
  for (int i = tid; i < V_; i += T) {
    unsigned long long mk = minkey[i];
    int p = i;
    if (mk != ~0ULL) {
      int sel = (int)(mk & 0xffffffffu);
      int a = comp[u[sel]], c = comp[v[sel]];
      p = (a == i) ? c : a;
      chosen[sel] = 1;
      myAny = 1;
    }
    p0[i] = p;
  }
  if (myAny) sAny = 1;
  __syncthreads();
  if (!sAny) {                       // no active merges anywhere: batch converged
    if (tid == 0) done[b] = 1;       // minkey already all-MAX; comp/chosen final
    return;
  }

  // break 2-cycles: smaller index becomes root (reference tie-break)
  for (int i = tid; i < V_; i += T) {
    int p = p0[i]; int gp = p0[p];
    p1[i] = (gp == i && i < p) ? i : p;
  }
  __syncthreads();

  // pointer jumping to fixpoint (early exit when parent[parent] == parent)
  int* cur = p1; int* alt = p0;
  for (int it = 0; it < JUMPS; ++it) {
    if (tid == 0) sChanged = 0;
    __syncthreads();
    int ch = 0;
    for (int i = tid; i < V_; i += T) {
      int c0 = cur[i];
      int np = cur[c0];
      alt[i] = np;
      ch |= (np != c0);
    }
    if (ch) sChanged = 1;
    __syncthreads();
    { int* tmp = cur; cur = alt; alt = tmp; }
    if (!sChanged) break;
  }

  // comp = parent[comp]; reset minkey for next round
  for (int i = tid; i < V_; i += T) {
    comp[i] = cur[comp[i]];
    minkey[i] = ~0ULL;
  }
}

__global__ void scan1_kernel(const unsigned char* __restrict__ chosen,
                             int* __restrict__ bsums, int nb) {
  __shared__ int sh[256];
  int b = blockIdx.y, blk = blockIdx.x, tid = threadIdx.x;
  const unsigned char* ch = chosen + (size_t)b * E_;
  int base = blk * 1024 + tid * 4;
  int s = 0;
#pragma unroll
  for (int j = 0; j < 4; ++j) { int idx = base + j; if (idx < E_) s += ch[idx]; }
  sh[tid] = s; __syncthreads();
  for (int off = 1; off < 256; off <<= 1) {
    int x = (tid >= off) ? sh[tid - off] : 0; __syncthreads();
    sh[tid] += x; __syncthreads();
  }
  if (tid == 255) bsums[b * nb + blk] = sh[255];
}

__global__ void scan2_kernel(int* __restrict__ bsums, int nb) {
  int b = blockIdx.x;
  if (threadIdx.x == 0) {
    int acc = 0;
    for (int j = 0; j < nb; ++j) { int t = bsums[b * nb + j]; bsums[b * nb + j] = acc; acc += t; }
  }
}

__global__ void scatter_kernel(const unsigned char* __restrict__ chosen,
                               const int* __restrict__ bsums, int nb,
                               int* __restrict__ out) {
  __shared__ int sh[256];
  int b = blockIdx.y, blk = blockIdx.x, tid = threadIdx.x;
  const unsigned char* ch = chosen + (size_t)b * E_;
  int base = blk * 1024 + tid * 4;
  int f[4]; int s = 0;
#pragma unroll
  for (int j = 0; j < 4; ++j) {
    int idx = base + j; f[j] = (idx < E_) ? (int)ch[idx] : 0; s += f[j];
  }
  sh[tid] = s; __syncthreads();
  for (int off = 1; off < 256; off <<= 1) {
    int x = (tid >= off) ? sh[tid - off] : 0; __syncthreads();
    sh[tid] += x; __syncthreads();
  }
  int pos = bsums[b * nb + blk] + (sh[tid] - s);
  int* ob = out + (size_t)b * (V_ - 1);
#pragma unroll
  for (int j = 0; j < 4; ++j) {
    if (f[j]) { if (pos < V_ - 1) ob[pos] = base + j; pos++; }
  }
}

extern "C" void kernel_launch(void* const* d_in, const int* in_sizes, int n_in,
                              void* d_out, int out_size, void* d_ws, size_t ws_size,
                              hipStream_t stream) {
  (void)in_sizes; (void)n_in; (void)out_size; (void)ws_size;
  const float* g = (const float*)d_in[0];
  int* out = (int*)d_out;
  char* ws = (char*)d_ws;

  size_t off = 0;
  auto take = [&](size_t bytes) { size_t o = off; off += (bytes + 255) & ~(size_t)255; return o; };
  unsigned long long* wkey  = (unsigned long long*)(ws + take((size_t)B_ * E_ * 8));
  int* u    = (int*)(ws + take((size_t)E_ * 4));
  int* v    = (int*)(ws + take((size_t)E_ * 4));
  int* comp = (int*)(ws + take((size_t)B_ * V_ * 4));
  int* par  = (int*)(ws + take((size_t)B_ * V_ * 4));
  int* par2 = (int*)(ws + take((size_t)B_ * V_ * 4));
  unsigned long long* minkey = (unsigned long long*)(ws + take((size_t)B_ * V_ * 8));
  unsigned char* chosen = (unsigned char*)(ws + take((size_t)B_ * E_));
  const int nb = (E_ + 1023) / 1024;
  int* bsums = (int*)(ws + take((size_t)B_ * nb * 4));
  int* done  = (int*)(ws + take((size_t)B_ * 4));

  setup_uv_kernel<<<(E_ + 255) / 256, 256, 0, stream>>>(u, v);
  init_kernel<<<1024, 256, 0, stream>>>(comp, minkey, chosen, out, done);

  {
    int wavesPerBatch = (E_ + 15) / 16;
    dim3 grid((wavesPerBatch + 7) / 8, B_);
    weight_wmma_kernel<<<grid, 256, 0, stream>>>(g, wkey);
  }

  for (int r = 0; r < ROUNDS; ++r) {
    dim3 rg((E_ + 255) / 256, B_);
    relax_kernel<<<rg, 256, 0, stream>>>(u, v, wkey, comp, minkey, done);
    merge_kernel<<<B_, 1024, 0, stream>>>(u, v, minkey, comp, par, par2, chosen, done);
  }

  dim3 sg(nb, B_);
  scan1_kernel<<<sg, 256, 0, stream>>>(chosen, bsums, nb);
  scan2_kernel<<<B_, 32, 0, stream>>>(bsums, nb);
  scatter_kernel<<<sg, 256, 0, stream>>>(chosen, bsums, nb, out);
}